// NODEQuantizer_27453430956159
// MI455X (gfx1250) — compile-verified
//
#include <hip/hip_runtime.h>
#include <hip/hip_bf16.h>
#include <stdint.h>

#define FDIM   256
#define NK     10
#define NSEG   9
#define NTIMES 10

typedef unsigned int u32x4 __attribute__((ext_vector_type(4)));
typedef int          i32x4 __attribute__((ext_vector_type(4)));
typedef int          i32x8 __attribute__((ext_vector_type(8)));

__device__ __forceinline__ uint32_t lds_byte_off(const void* p) {
    // Low 32 bits of a generic pointer to LDS are the LDS byte offset
    // (flat->LDS aperture mapping truncates to addr[31:0]).
    return (uint32_t)(uintptr_t)p;
}

__global__ __launch_bounds__(FDIM) void node_spline_kernel(
    const float* __restrict__ x, const float* __restrict__ weights,
    float* __restrict__ out, int rows) {
    __shared__ float  sKnots[FDIM * NK];          // 10240 B
    __shared__ float4 sA[FDIM * NSEG];            // 36864 B (k, y, b, c)
    __shared__ float  sD[FDIM * NSEG];            //  9216 B (d)

    // ---- Stage knot table global -> LDS via Tensor Data Mover (gfx1250) ----
#if defined(__HIP_DEVICE_COMPILE__) && defined(__gfx1250__) && \
    __has_builtin(__builtin_amdgcn_tensor_load_to_lds) && \
    __has_builtin(__builtin_amdgcn_s_wait_tensorcnt)
    if (threadIdx.x < 32u) {   // wave 0 only (uniform branch, wave32)
        const uint64_t ga = (uint64_t)(uintptr_t)weights;
        u32x4 g0;
        g0[0] = 1u;                                 // count=1, user D#
        g0[1] = lds_byte_off(sKnots);               // LDS byte address
        g0[2] = (uint32_t)ga;                       // global_addr[31:0]
        g0[3] = (uint32_t)((ga >> 32) & 0x01FFFFFFu) | 0x80000000u; // [56:32] | type=2
        i32x8 g1;
        g1[0] = (int)(2u << 16);                    // data_size = 4B
        g1[1] = (int)(((uint32_t)(FDIM * NK) & 0xFFFFu) << 16); // tensor_dim0 lo16
        g1[2] = (int)(1u << 16);                    // tensor_dim0 hi=0 | tensor_dim1=1
        g1[3] = (int)((uint32_t)(FDIM * NK) << 16); // tile_dim0 = 2560
        g1[4] = 1;                                  // tile_dim1 = 1, tile_dim2 = 0
        g1[5] = FDIM * NK;                          // tensor_dim0_stride
        g1[6] = 0;
        g1[7] = 0;
        i32x4 gz = {0, 0, 0, 0};
#if __clang_major__ >= 23
        i32x8 gz8 = {0, 0, 0, 0, 0, 0, 0, 0};
        __builtin_amdgcn_tensor_load_to_lds(g0, g1, gz, gz, gz8, 0);
#else
        __builtin_amdgcn_tensor_load_to_lds(g0, g1, gz, gz, 0);
#endif
        __builtin_amdgcn_s_wait_tensorcnt(0);
    }
#else
    for (int t = (int)threadIdx.x; t < FDIM * NK; t += FDIM)
        sKnots[t] = weights[t];
#endif
    __syncthreads();

    // ---- Per-feature natural cubic spline coefficients (Thomas, 8x8) ----
    const int f = (int)threadIdx.x;
    float kn[NK];
#pragma unroll
    for (int j = 0; j < NK; ++j) kn[j] = sKnots[f * NK + j];

    float h[NSEG], sl[NSEG];
#pragma unroll
    for (int j = 0; j < NSEG; ++j) {
        const float y0 = (j & 1) ? -1.0f : 1.0f;
        h[j]  = kn[j + 1] - kn[j];
        sl[j] = (-y0 - y0) / h[j];      // (y[j+1]-y[j])/h
    }
    float cp[7], dp[8];
    {
        const float diag0 = 2.0f * (h[0] + h[1]);
        cp[0] = h[1] / diag0;
        dp[0] = 6.0f * (sl[1] - sl[0]) / diag0;
#pragma unroll
        for (int i = 1; i < 7; ++i) {
            const float den = 2.0f * (h[i] + h[i + 1]) - h[i] * cp[i - 1];
            cp[i] = h[i + 1] / den;
            dp[i] = (6.0f * (sl[i + 1] - sl[i]) - h[i] * dp[i - 1]) / den;
        }
        const float den7 = 2.0f * (h[7] + h[8]) - h[7] * cp[6];
        dp[7] = (6.0f * (sl[8] - sl[7]) - h[7] * dp[6]) / den7;
    }
    float M[NK];
    M[0] = 0.0f; M[NK - 1] = 0.0f; M[8] = dp[7];
#pragma unroll
    for (int i = 6; i >= 0; --i) M[i + 1] = dp[i] - cp[i] * M[i + 2];

    const int base = f * NSEG;
#pragma unroll
    for (int i = 0; i < NSEG; ++i) {
        const float yi = (i & 1) ? -1.0f : 1.0f;
        const float ci = 0.5f * M[i];
        const float di = (M[i + 1] - M[i]) / (6.0f * h[i]);
        const float bi = sl[i] - h[i] * (2.0f * M[i] + M[i + 1]) * (1.0f / 6.0f);
        sA[base + i] = make_float4(kn[i], yi, bi, ci);
        sD[base + i] = di;
    }
    __syncthreads();

    // ---- Spline eval: branchless search + conflict-free LDS gather ----
    auto evalS = [&](float v) -> float {
        int i = 0;
#pragma unroll
        for (int j = 1; j < NSEG; ++j) i += (v >= kn[j]) ? 1 : 0;
        const float4 a = sA[base + i];   // ds_load_b128
        const float  d = sD[base + i];   // ds_load_b32
        const float fr = v - a.x;
        return fmaf(fr, fmaf(fr, fmaf(fr, d, a.w), a.z), a.y);
    };

    // ---- RK 3/8-rule integration, 2 rows per iteration for ILP ----
    const float dt = 1.0f / (float)(NTIMES - 1);
    for (int r = (int)blockIdx.x * 2; r < rows; r += (int)gridDim.x * 2) {
        const int   r1   = r + 1;
        const bool  has2 = (r1 < rows);
        const size_t e0 = (size_t)r  * FDIM + (size_t)f;
        const size_t e1 = (size_t)r1 * FDIM + (size_t)f;
        float s0 = x[e0];
        float s1 = has2 ? x[e1] : 0.5f;
#pragma unroll 1
        for (int st = 0; st < NTIMES - 1; ++st) {
            const float k1a = evalS(s0);
            const float k1b = evalS(s1);
            const float k2a = evalS(fmaf(dt * (1.0f / 3.0f), k1a, s0));
            const float k2b = evalS(fmaf(dt * (1.0f / 3.0f), k1b, s1));
            const float k3a = evalS(fmaf(dt, k2a - k1a * (1.0f / 3.0f), s0));
            const float k3b = evalS(fmaf(dt, k2b - k1b * (1.0f / 3.0f), s1));
            const float k4a = evalS(fmaf(dt, k1a - k2a + k3a, s0));
            const float k4b = evalS(fmaf(dt, k1b - k2b + k3b, s1));
            s0 = fmaf(dt * 0.125f, k1a + 3.0f * (k2a + k3a) + k4a, s0);
            s1 = fmaf(dt * 0.125f, k1b + 3.0f * (k2b + k3b) + k4b, s1);
        }
        out[e0] = fminf(fmaxf(s0, 0.0f), 1.0f);
        if (has2) out[e1] = fminf(fmaxf(s1, 0.0f), 1.0f);
    }
}

extern "C" void kernel_launch(void* const* d_in, const int* in_sizes, int n_in,
                              void* d_out, int out_size, void* d_ws, size_t ws_size,
                              hipStream_t stream) {
    (void)n_in; (void)out_size; (void)d_ws; (void)ws_size;
    const float* x = (const float*)d_in[0];
    const float* w = (const float*)d_in[1];
    float* out = (float*)d_out;
    const int rows = in_sizes[0] / FDIM;

    int blocks = (rows + 1) / 2;
    if (blocks > 2048) blocks = 2048;   // grid-stride; keeps TDM staging ~once per WGP slot
    if (blocks < 1) blocks = 1;
    node_spline_kernel<<<blocks, FDIM, 0, stream>>>(x, w, out, rows);
}